// MertonJumpDiffusion_67869073211470
// MI455X (gfx1250) — compile-verified
//
#include <hip/hip_runtime.h>
#include <hip/hip_bf16.h>

#if defined(__has_builtin)
#if __has_builtin(__builtin_amdgcn_global_load_async_to_lds_b128)
#define HAVE_ASYNC 1
#endif
#endif
#ifndef HAVE_ASYNC
#define HAVE_ASYNC 0
#endif

#if HAVE_ASYNC && defined(__has_builtin)
#if __has_builtin(__builtin_amdgcn_s_wait_asynccnt)
#define WAIT_ASYNC(n) __builtin_amdgcn_s_wait_asynccnt(n)
#else
#define WAIT_ASYNC(n) asm volatile("s_wait_asynccnt " #n ::: "memory")
#endif
#endif

typedef __attribute__((ext_vector_type(2))) float v2f;
typedef __attribute__((ext_vector_type(4))) float v4f;
typedef __attribute__((ext_vector_type(8))) float v8f;
typedef __attribute__((ext_vector_type(4))) int   v4i;

#define N_STEPS   2048
#define N_PATHS   32768
#define OUT_COLS  (N_STEPS + 1)
#define DT_F      (1.0f / 2048.0f)
#define SIGMA_F   0.2f
#define SIGMA_J_F 0.3f
#define KAPPA_F   0.046027859f   /* exp(0.045) - 1 */
#define LDS_STRIDE 20            /* 16 steps + pad -> 80B rows, 16B aligned */

#if HAVE_ASYNC
typedef __attribute__((address_space(1))) v4i* gv4i_p;  // global int4*
typedef __attribute__((address_space(3))) v4i* lv4i_p;  // LDS int4*
__device__ __forceinline__ void async_cp16(const void* g, void* l) {
    // GLOBAL_LOAD_ASYNC_TO_LDS_B128: 16B per lane, tracked by ASYNCcnt
    __builtin_amdgcn_global_load_async_to_lds_b128((gv4i_p)g, (lv4i_p)l, 0, 0);
}
#endif

// One wave (32 lanes) owns 16 paths across all 2048 steps.
// Per 16-step tile: double-buffered async Global->LDS staging (6x b128/tile),
// B-layout gather from LDS, prefix-sum via 4 chained v_wmma_f32_16x16x4_f32
// against lower-triangular-ones A chunks (carry pre-loaded in C), then exp()
// and an LDS transpose for coalesced row-major output stores.
__global__ __launch_bounds__(256)
void merton_scan_wmma(const float* __restrict__ S0,
                      const float* __restrict__ zdiff,
                      const float* __restrict__ zjump,
                      const int*   __restrict__ njump,
                      float*       __restrict__ out)
{
    // [wave][buf][array][row*16+path] : 8*2*3*256 floats = 48 KB
    __shared__ __align__(16) float stage[8 * 2 * 3 * 256];
    __shared__ __align__(16) float xpose[8][16 * LDS_STRIDE];

    const int lane = threadIdx.x & 31;
    const int wave = threadIdx.x >> 5;
    const int half = lane >> 4;      // 0: lanes 0-15, 1: lanes 16-31
    const int n16  = lane & 15;      // N index / path-in-tile
    const int p0   = (blockIdx.x * 8 + wave) * 16;

    const float s0 = S0[0];

    // A = lower-triangular ones, 4 chunks of 16x4 (f32 A layout: K = v + 2*half)
    v2f a[4];
#pragma unroll
    for (int c = 0; c < 4; ++c) {
        a[c].x = (4 * c + 0 + 2 * half <= n16) ? 1.0f : 0.0f;
        a[c].y = (4 * c + 1 + 2 * half <= n16) ? 1.0f : 0.0f;
    }

    // Column 0 of the output: S0 for every path.
    if (lane < 16) out[(size_t)(p0 + lane) * OUT_COLS] = s0;

    const float cdiff = SIGMA_F * __fsqrt_rn(DT_F);  // sigma*sqrt(dt)
    const float kd    = KAPPA_F * DT_F;

    float carry = 0.0f;  // running log-path for path p0 + n16

#if HAVE_ASYNC
    const int r4  = lane >> 2;        // staging: row-within-octet 0..7
    const int c16 = (lane & 3) * 4;   // staging: 4-path column 0,4,8,12
    auto issue_tile = [&](int tb, int buf) {
        const size_t colbase = (size_t)p0 + (size_t)c16;
        float* dst = &stage[(size_t)((wave * 2 + buf) * 3) * 256];
#pragma unroll
        for (int i = 0; i < 2; ++i) {
            const int    row  = i * 8 + r4;
            const size_t goff = (size_t)(tb + row) * N_PATHS + colbase;
            const int    loff = row * 16 + c16;
            async_cp16(zdiff + goff,               dst + 0 * 256 + loff);
            async_cp16(zjump + goff,               dst + 1 * 256 + loff);
            async_cp16((const float*)njump + goff, dst + 2 * 256 + loff);
        }
    };
    int cur = 0;
    issue_tile(0, 0);
#endif

    for (int base = 0; base < N_STEPS; base += 16) {
#if HAVE_ASYNC
        if (base + 16 < N_STEPS) {
            issue_tile(base + 16, cur ^ 1);   // prefetch next tile (6 async b128)
            WAIT_ASYNC(6);                    // current tile resident (in-order)
        } else {
            WAIT_ASYNC(0);
        }
        asm volatile("" ::: "memory");
        const float* sb = &stage[(size_t)((wave * 2 + cur) * 3) * 256];
#endif

        // ---- gather 16x16 tile of log-returns in B (4x16 chunk) layout ----
        float xb[8];
#pragma unroll
        for (int u = 0; u < 8; ++u) {
            const int tl = 4 * (u >> 1) + (u & 1) + 2 * half;  // K within tile
            const int t  = base + tl;
            const float lam = 0.1f + 0.9f * __expf(-0.01f * DT_F * (float)t);
#if HAVE_ASYNC
            const int   off = tl * 16 + n16;
            const float zdv = sb[0 * 256 + off];
            const float zjv = sb[1 * 256 + off];
            const float nj  = (float)((const int*)sb)[2 * 256 + off];
#else
            const size_t idx = (size_t)t * N_PATHS + (size_t)(p0 + n16);
            const float zdv = zdiff[idx];
            const float zjv = zjump[idx];
            const float nj  = (float)njump[idx];
#endif
            xb[u] = -lam * kd + cdiff * zdv + __fsqrt_rn(nj) * SIGMA_J_F * zjv;
        }

        // ---- Y = L*X + carry  via 4 chained f32 WMMAs ----
        v8f d;
#pragma unroll
        for (int r = 0; r < 8; ++r) d[r] = carry;   // C[i][j] = carry[j]
#pragma unroll
        for (int c = 0; c < 4; ++c) {
            v2f b;
            b.x = xb[2 * c + 0];
            b.y = xb[2 * c + 1];
            d = __builtin_amdgcn_wmma_f32_16x16x4_f32(
                    /*neg_a=*/false, a[c], /*neg_b=*/false, b,
                    /*c_mod=*/(short)0, d, /*reuse_a=*/false, /*reuse_b=*/false);
        }

        // New carry = row M=15 of D: VGPR7, lanes 16-31, N = lane-16.
        carry = __shfl(d[7], 16 + n16, 32);

        // ---- exp, LDS transpose (per-wave private region) ----
        v4f w0, w1;
#pragma unroll
        for (int r = 0; r < 4; ++r) w0[r] = s0 * __expf(d[r]);
#pragma unroll
        for (int r = 0; r < 4; ++r) w1[r] = s0 * __expf(d[r + 4]);

        // lane holds path n16, steps M = 8*half .. 8*half+7 (consecutive)
        float* rowp = &xpose[wave][n16 * LDS_STRIDE + 8 * half];
        *(v4f*)(rowp)     = w0;
        *(v4f*)(rowp + 4) = w1;

        __syncthreads();

        // ---- coalesced writeback: half-wave per path row, 64B segments ----
#pragma unroll
        for (int rr = 0; rr < 8; ++rr) {
            const int prow = 2 * rr + half;
            const float v  = xpose[wave][prow * LDS_STRIDE + n16];
            out[(size_t)(p0 + prow) * OUT_COLS + (size_t)(base + 1 + n16)] = v;
        }

        __syncthreads();

#if HAVE_ASYNC
        cur ^= 1;
#endif
    }
}

extern "C" void kernel_launch(void* const* d_in, const int* in_sizes, int n_in,
                              void* d_out, int out_size, void* d_ws, size_t ws_size,
                              hipStream_t stream) {
    const float* S0 = (const float*)d_in[0];
    const float* zd = (const float*)d_in[1];
    const float* zj = (const float*)d_in[2];
    const int*   nj = (const int*)d_in[3];
    float*       out = (float*)d_out;

    dim3 block(256);                      // 8 wave32s
    dim3 grid(N_PATHS / (8 * 16));        // 256 blocks, 16 paths per wave
    hipLaunchKernelGGL(merton_scan_wmma, grid, block, 0, stream,
                       S0, zd, zj, nj, out);
}